// CovariantGaugeAdapter_68221260530105
// MI455X (gfx1250) — compile-verified
//
#include <hip/hip_runtime.h>

#define SS 512
#define DD 768
#define NH 12
#define HD 64
#define RK 16
#define LN_EPS 1e-5f

typedef __attribute__((ext_vector_type(2))) float v2f;
typedef __attribute__((ext_vector_type(8))) float v8f;

#if __has_builtin(__builtin_amdgcn_tanhf)
__device__ __forceinline__ float my_tanh(float x) {
  return __builtin_amdgcn_tanhf(x);   // native v_tanh_f32 on gfx1250
}
#else
__device__ __forceinline__ float my_tanh(float x) {
  // tanh(x) = 1 - 2/(exp(2x)+1); exact limits at +/-inf via exp2 overflow/underflow
  float e = __builtin_amdgcn_exp2f(x * 2.885390081777927f); // 2*log2(e)
  return 1.0f - 2.0f * __builtin_amdgcn_rcpf(e + 1.0f);
}
#endif

__device__ __forceinline__ float fast_silu(float x) {
  float e = __builtin_amdgcn_exp2f(-x * 1.44269504088896341f);
  return x * __builtin_amdgcn_rcpf(1.0f + e);
}

// ---------------- Stage 1: LayerNorm + field generator ----------------
// grid: (S) blocks x 256 threads. Produces A_q, A_k as [H][S][HD], A_v as [S][D].
__global__ __launch_bounds__(256) void stage1_ln_fieldgen(
    const float* __restrict__ hs, const float* __restrict__ gamma,
    const float* __restrict__ beta, const float* __restrict__ W1,
    const float* __restrict__ W2, float* __restrict__ Aq,
    float* __restrict__ Ak, float* __restrict__ Av) {
  __shared__ float xs[DD];
  __shared__ float hval[RK];
  __shared__ float red[16];
  __shared__ float mu_s, rstd_s;

  const int s = blockIdx.x;
  const int t = threadIdx.x;
  const float* xrow = hs + (size_t)s * DD;

  float xloc[3];
  float lsum = 0.f, lsq = 0.f;
  #pragma unroll
  for (int k = 0; k < 3; ++k) {
    float v = xrow[t + 256 * k];
    xloc[k] = v;
    lsum += v;
    lsq += v * v;
  }
  #pragma unroll
  for (int off = 16; off >= 1; off >>= 1) {
    lsum += __shfl_down(lsum, off, 32);
    lsq  += __shfl_down(lsq, off, 32);
  }
  const int wv = t >> 5, ln = t & 31;
  if (ln == 0) { red[wv] = lsum; red[8 + wv] = lsq; }
  __syncthreads();
  if (t == 0) {
    float sm = 0.f, sq = 0.f;
    #pragma unroll
    for (int w = 0; w < 8; ++w) { sm += red[w]; sq += red[8 + w]; }
    float mu = sm * (1.0f / DD);
    float var = sq * (1.0f / DD) - mu * mu;
    mu_s = mu;
    rstd_s = __builtin_amdgcn_rsqf(var + LN_EPS);
  }
  __syncthreads();
  const float mu = mu_s, rstd = rstd_s;
  #pragma unroll
  for (int k = 0; k < 3; ++k) {
    int e = t + 256 * k;
    xs[e] = (xloc[k] - mu) * rstd * gamma[e] + beta[e];
  }
  __syncthreads();

  // h[r] = silu(xs . W1[r,:]) : 16 threads per r (sub-groups of a wave32)
  {
    const int r = t >> 4;      // 0..15
    const int l = t & 15;
    const float* w1r = W1 + r * DD;
    float acc = 0.f;
    for (int e = l; e < DD; e += 16) acc += xs[e] * w1r[e];
    #pragma unroll
    for (int off = 8; off >= 1; off >>= 1) acc += __shfl_down(acc, off, 16);
    if (l == 0) hval[r] = fast_silu(acc);
  }
  __syncthreads();

  float hreg[RK];
  #pragma unroll
  for (int r = 0; r < RK; ++r) hreg[r] = hval[r];

  // A[j] = sum_r h[r] * W2[j][r], j in [0, 3D)
  #pragma unroll
  for (int jj = 0; jj < 9; ++jj) {
    int j = t + 256 * jj;
    const float* w2r = W2 + (size_t)j * RK;
    float acc = 0.f;
    #pragma unroll
    for (int r = 0; r < RK; ++r) acc += hreg[r] * w2r[r];
    if (j < DD) {
      int hh = j >> 6, d = j & 63;
      Aq[((size_t)hh * SS + s) * HD + d] = acc;
    } else if (j < 2 * DD) {
      int j2 = j - DD;
      int hh = j2 >> 6, d = j2 & 63;
      Ak[((size_t)hh * SS + s) * HD + d] = acc;
    } else {
      Av[(size_t)s * DD + (j - 2 * DD)] = acc;
    }
  }
}

// ---------------- Stage 1b: per-head 64x64 projection ----------------
// grid: (S/4, H) x 256 threads. Aout[h,s,d] = sum_e Win[d,e] * Ain[h,s,e]
__global__ __launch_bounds__(256) void stage1b_proj(
    const float* __restrict__ Win, const float* __restrict__ Ain,
    float* __restrict__ Aout) {
  __shared__ float W[HD * HD];
  const int t = threadIdx.x;
  #pragma unroll
  for (int k = t; k < HD * HD; k += 256) W[k] = Win[k];
  __syncthreads();
  const int h = blockIdx.y;
  const int s = blockIdx.x * 4 + (t >> 6);
  const int d = t & 63;
  const float* arow = Ain + ((size_t)h * SS + s) * HD;
  const float* wrow = W + d * HD;
  float acc = 0.f;
  #pragma unroll
  for (int e = 0; e < HD; ++e) acc += wrow[e] * arow[e];
  Aout[((size_t)h * SS + s) * HD + d] = acc;
}

// ---------------- Stage 2: gauge_bias tiles (WMMA + tanh) ----------------
// grid: (S/16, S/16, H), 1 wave (32 threads) per 16x16 tile.
#define PAD (HD + 4)   // float4-aligned rows, conflict-free LDS
__global__ __launch_bounds__(32) void stage2_gauge(
    const float* __restrict__ q_base, const float* __restrict__ k_base,
    const float* __restrict__ Aq, const float* __restrict__ Ak,
    const float* __restrict__ Aqp, const float* __restrict__ Akp,
    const float* __restrict__ rbv, const float* __restrict__ g_attn,
    const float* __restrict__ g_rel, float* __restrict__ out) {
  __shared__ __align__(16) float q_lds[16][PAD];  // A_q rows (i)
  __shared__ __align__(16) float k_lds[16][PAD];  // A_k rows (j)
  __shared__ __align__(16) float rb[HD];

  const int L = threadIdx.x;
  const int i0 = blockIdx.x * 16;
  const int j0 = blockIdx.y * 16;
  const int h = blockIdx.z;

  // coalesced float4 tile loads: 16 rows x 16 float4 = 256 float4, 32 lanes
  for (int idx = L; idx < 16 * (HD / 4); idx += 32) {
    const int rr = idx >> 4, cc = (idx & 15) * 4;
    *(float4*)&q_lds[rr][cc] =
        *(const float4*)&Aq[((size_t)h * SS + i0 + rr) * HD + cc];
    *(float4*)&k_lds[rr][cc] =
        *(const float4*)&Ak[((size_t)h * SS + j0 + rr) * HD + cc];
  }
  rb[L] = rbv[h * HD + L];
  rb[L + 32] = rbv[h * HD + L + 32];
  __syncthreads();

  const int hf = L >> 4;   // half-wave select
  const int lp = L & 15;

  const float* qA = q_base + ((size_t)h * SS + i0 + lp) * HD;  // A rows for b1
  const float* aK = Akp    + ((size_t)h * SS + j0 + lp) * HD;  // B cols for b1
  const float* aQ = Aqp    + ((size_t)h * SS + i0 + lp) * HD;  // A rows for b2
  const float* kB = k_base + ((size_t)h * SS + j0 + lp) * HD;  // B cols for b2

  v8f c1 = {};
  v8f c2 = {};
  #pragma unroll
  for (int kk = 0; kk < HD; kk += 4) {
    const int k = kk + hf * 2;
    v2f a1 = *(const v2f*)(qA + k);
    v2f b1 = *(const v2f*)(aK + k);
    c1 = __builtin_amdgcn_wmma_f32_16x16x4_f32(false, a1, false, b1,
                                               (short)0, c1, false, false);
    v2f a2 = *(const v2f*)(aQ + k);
    v2f b2 = *(const v2f*)(kB + k);
    c2 = __builtin_amdgcn_wmma_f32_16x16x4_f32(false, a2, false, b2,
                                               (short)0, c2, false, false);
  }

  // b3: col jl = lp is the same for all 8 C-rows this lane owns, so k/rb
  // loads are hoisted out of the r loop; q-row reads are half-wave broadcasts.
  float acc[8] = {0.f, 0.f, 0.f, 0.f, 0.f, 0.f, 0.f, 0.f};
  for (int dd = 0; dd < HD; dd += 4) {
    const float4 kv = *(const float4*)&k_lds[lp][dd];
    const float4 rv = *(const float4*)&rb[dd];
    #pragma unroll
    for (int r = 0; r < 8; ++r) {
      const float4 qv = *(const float4*)&q_lds[r + hf * 8][dd];
      acc[r] += my_tanh(kv.x - qv.x) * rv.x;
      acc[r] += my_tanh(kv.y - qv.y) * rv.y;
      acc[r] += my_tanh(kv.z - qv.z) * rv.z;
      acc[r] += my_tanh(kv.w - qv.w) * rv.w;
    }
  }

  const float ga = g_attn[h] * 0.125f;  // scale = 1/sqrt(64)
  const float gr = g_rel[h];

  #pragma unroll
  for (int r = 0; r < 8; ++r) {
    const int il = r + hf * 8;   // row in C layout
    float val = ga * (c1[r] + c2[r]) + gr * acc[r];
    out[((size_t)(h * SS + i0 + il)) * SS + j0 + lp] = val;
  }
}

// ---------------- Stage 3: delta_v = tanh(g_val) * (A_v @ Wv.T) ----------------
// grid: (S/16, D/16), 1 wave per 16x16 tile, K = 768.
__global__ __launch_bounds__(32) void stage3_deltav(
    const float* __restrict__ Av, const float* __restrict__ Wv,
    const float* __restrict__ g_val, float* __restrict__ out) {
  const int L = threadIdx.x;
  const int m0 = blockIdx.x * 16;  // token tile
  const int n0 = blockIdx.y * 16;  // feature tile
  const int hf = L >> 4, lp = L & 15;

  const float* arow = Av + (size_t)(m0 + lp) * DD;
  const float* wrow = Wv + (size_t)(n0 + lp) * DD;  // B[e][n] = Wv[n][e]

  v8f c = {};
  for (int kk = 0; kk < DD; kk += 4) {
    const int k = kk + hf * 2;
    v2f a = *(const v2f*)(arow + k);
    v2f b = *(const v2f*)(wrow + k);
    c = __builtin_amdgcn_wmma_f32_16x16x4_f32(false, a, false, b,
                                              (short)0, c, false, false);
  }
  const float tg = my_tanh(g_val[n0 + lp]);
  #pragma unroll
  for (int r = 0; r < 8; ++r) {
    const int mm = m0 + r + hf * 8;
    out[(size_t)mm * DD + n0 + lp] = c[r] * tg;
  }
}

extern "C" void kernel_launch(void* const* d_in, const int* in_sizes, int n_in,
                              void* d_out, int out_size, void* d_ws,
                              size_t ws_size, hipStream_t stream) {
  const float* hs    = (const float*)d_in[0];   // [1,S,D]
  const float* qb    = (const float*)d_in[1];   // [1,H,S,HD]
  const float* kb    = (const float*)d_in[2];   // [1,H,S,HD]
  const float* gam   = (const float*)d_in[3];   // [D]
  const float* bet   = (const float*)d_in[4];   // [D]
  const float* W1    = (const float*)d_in[5];   // [R,D]
  const float* W2    = (const float*)d_in[6];   // [3D,R]
  const float* Wq    = (const float*)d_in[7];   // [HD,HD]
  const float* Wk    = (const float*)d_in[8];   // [HD,HD]
  const float* Wv    = (const float*)d_in[9];   // [D,D]
  const float* rbv   = (const float*)d_in[10];  // [H,HD]
  const float* ga    = (const float*)d_in[11];  // [H]
  const float* gr    = (const float*)d_in[12];  // [H]
  const float* gv    = (const float*)d_in[13];  // [D]

  const size_t NHS = (size_t)NH * SS * HD;  // 393216
  float* ws  = (float*)d_ws;
  float* Aq  = ws;
  float* Ak  = ws + NHS;
  float* Aqp = ws + 2 * NHS;
  float* Akp = ws + 3 * NHS;
  float* Av  = ws + 4 * NHS;

  float* gauge = (float*)d_out;                       // [H,S,S]
  float* dv    = gauge + (size_t)NH * SS * SS;        // [S,D]

  stage1_ln_fieldgen<<<dim3(SS), dim3(256), 0, stream>>>(hs, gam, bet, W1, W2,
                                                         Aq, Ak, Av);
  stage1b_proj<<<dim3(SS / 4, NH), dim3(256), 0, stream>>>(Wq, Aq, Aqp);
  stage1b_proj<<<dim3(SS / 4, NH), dim3(256), 0, stream>>>(Wk, Ak, Akp);
  stage2_gauge<<<dim3(SS / 16, SS / 16, NH), dim3(32), 0, stream>>>(
      qb, kb, Aq, Ak, Aqp, Akp, rbv, ga, gr, gauge);
  stage3_deltav<<<dim3(SS / 16, DD / 16), dim3(32), 0, stream>>>(Av, Wv, gv, dv);
}